// TensorProduct_4621384810524
// MI455X (gfx1250) — compile-verified
//
#include <hip/hip_runtime.h>
#include <stdint.h>

typedef float v2f __attribute__((ext_vector_type(2)));
typedef float v4f __attribute__((ext_vector_type(4)));
typedef float v8f __attribute__((ext_vector_type(8)));
typedef unsigned int v4u __attribute__((ext_vector_type(4)));

#define U_DIM   128
#define NSEG0   32
#define NSEG1   8
#define NSEGO   32
#define NPATH   128
#define Z_DIM   16384
#define ROWS_PER_BLOCK 16
#define NBLOCKS (Z_DIM / ROWS_PER_BLOCK)
#define X0_ROW  (NSEG0 * U_DIM)          /* 4096 floats */
#define X1_ROW  (NSEG1 * U_DIM)          /* 1024 floats */
#define ROW_FLOATS (X0_ROW + X1_ROW)     /* 5120 floats = 20 KB */

// Async global->LDS copy of 16B per lane (512B per wave-instruction).
// GV addressing mode: 64-bit VGPR address, saddr = off. Tracked by ASYNCcnt.
__device__ __forceinline__ void async_copy16(uint32_t lds_byte_addr, const void* gaddr) {
  asm volatile("global_load_async_to_lds_b128 %0, %1, off"
               :: "v"(lds_byte_addr), "v"(gaddr)
               : "memory");
}

__global__ __launch_bounds__(256)
void tp_wmma_kernel(const float* __restrict__ x0,
                    const float* __restrict__ x1,
                    const float* __restrict__ coeffs,
                    const int*   __restrict__ idx0,
                    const int*   __restrict__ idx1,
                    const int*   __restrict__ idx_out,
                    float*       __restrict__ out)
{
  // Double-buffered staged rows: [0..4095] = x0 row, [4096..5119] = x1 row.
  __shared__ __align__(16) float xbuf[2][ROW_FLOATS];
  // A-matrix in WMMA fragment order: atab[kk*32+lane] =
  //   { A[n,p0], A[n,p0+1], A[n+16,p0], A[n+16,p0+1] },  p0 = 4*kk + 2*(lane>>4)
  __shared__ v4f atab[1024];
  // Gather offsets (element units): pinfo[p>>1] =
  //   { idx0[p]*128, 4096+idx1[p]*128, idx0[p+1]*128, 4096+idx1[p+1]*128 }
  __shared__ v4u pinfo[64];

  const int tid  = threadIdx.x;
  const int lane = tid & 31;
  const int wv   = tid >> 5;       // wave 0..7 -> u-tile
  const int nn   = lane & 15;      // N within tile / M within tile
  const int hb   = lane >> 4;      // lane half
  const int u    = wv * 16 + nn;   // this lane's channel

  // ---- build constant tables (once per block, reused for 16 rows) ----
  for (int e = tid; e < 1024; e += 256) {
    const int kk = e >> 5, l = e & 31;
    const int en = l & 15, eh = l >> 4;
    const int p0 = 4 * kk + 2 * eh;
    const int io0 = idx_out[p0], io1 = idx_out[p0 + 1];
    const float c0 = coeffs[p0], c1 = coeffs[p0 + 1];
    v4f a;
    a[0] = (io0 == en)      ? c0 : 0.0f;
    a[1] = (io1 == en)      ? c1 : 0.0f;
    a[2] = (io0 == en + 16) ? c0 : 0.0f;
    a[3] = (io1 == en + 16) ? c1 : 0.0f;
    atab[e] = a;
  }
  if (tid < 64) {
    const int p = 2 * tid;
    v4u pi;
    pi[0] = (unsigned)idx0[p] * U_DIM;
    pi[1] = (unsigned)(X0_ROW) + (unsigned)idx1[p] * U_DIM;
    pi[2] = (unsigned)idx0[p + 1] * U_DIM;
    pi[3] = (unsigned)(X0_ROW) + (unsigned)idx1[p + 1] * U_DIM;
    pinfo[tid] = pi;
  }

  const int z0 = blockIdx.x * ROWS_PER_BLOCK;

  // Issue async preload of row r into LDS buffer b: 5 async instrs per wave.
  auto preload = [&](int r, int b) {
    const float* s0 = x0 + (size_t)(z0 + r) * X0_ROW;
    const float* s1 = x1 + (size_t)(z0 + r) * X1_ROW;
    const uint32_t lbase = (uint32_t)(uintptr_t)&xbuf[b][0];
#pragma unroll
    for (int i = 0; i < 4; ++i) {
      const int e = (i * 256 + tid) * 4;                    // float index in x0 row
      async_copy16(lbase + (uint32_t)e * 4u, (const void*)(s0 + e));
    }
    const int e1 = tid * 4;                                 // float index in x1 row
    async_copy16(lbase + (uint32_t)(X0_ROW + e1) * 4u, (const void*)(s1 + e1));
  };

  preload(0, 0);
  preload(1, 1);

  for (int r = 0; r < ROWS_PER_BLOCK; ++r) {
    // Per-wave: rows r and r+1 may be in flight (5 async ops each).
    // Waiting for <=5 guarantees row r landed (async loads complete in order).
    if (r < ROWS_PER_BLOCK - 1) {
      asm volatile("s_wait_asynccnt 0x5" ::: "memory");
    } else {
      asm volatile("s_wait_asynccnt 0x0" ::: "memory");
    }
    __syncthreads();   // all waves' data for row r visible (also fences tables)

    const int b = r & 1;
    const float* xr = &xbuf[b][0];

    v8f c0 = {0.f, 0.f, 0.f, 0.f, 0.f, 0.f, 0.f, 0.f};  // o-tile 0 (rows 0..15)
    v8f c1 = {0.f, 0.f, 0.f, 0.f, 0.f, 0.f, 0.f, 0.f};  // o-tile 1 (rows 16..31)

#pragma unroll 4
    for (int kk = 0; kk < 32; ++kk) {
      const v4u pi = pinfo[2 * kk + hb];          // paths p0, p0+1 for this half
      const float x0a = xr[pi[0] + u];
      const float x1a = xr[pi[1] + u];
      const float x0b = xr[pi[2] + u];
      const float x1b = xr[pi[3] + u];
      v2f bf;
      bf[0] = x0a * x1a;                          // B[p0,   u]
      bf[1] = x0b * x1b;                          // B[p0+1, u]
      const v4f af = atab[kk * 32 + lane];
      v2f a0; a0[0] = af[0]; a0[1] = af[1];
      v2f a1; a1[0] = af[2]; a1[1] = af[3];
      c0 = __builtin_amdgcn_wmma_f32_16x16x4_f32(false, a0, false, bf,
                                                 (short)0, c0, false, false);
      c1 = __builtin_amdgcn_wmma_f32_16x16x4_f32(false, a1, false, bf,
                                                 (short)0, c1, false, false);
    }

    // D layout: VGPR j -> rows {j, j+8} (lane halves), N = nn.
    float* orow = out + (size_t)(z0 + r) * (NSEGO * U_DIM);
    const int ob = hb * 8;
#pragma unroll
    for (int j = 0; j < 8; ++j)
      orow[(ob + j) * U_DIM + u] = c0[j];
#pragma unroll
    for (int j = 0; j < 8; ++j)
      orow[(16 + ob + j) * U_DIM + u] = c1[j];

    __syncthreads();   // everyone done reading xbuf[b] before it is refilled
    if (r + 2 < ROWS_PER_BLOCK) preload(r + 2, b);
  }
}

extern "C" void kernel_launch(void* const* d_in, const int* in_sizes, int n_in,
                              void* d_out, int out_size, void* d_ws, size_t ws_size,
                              hipStream_t stream) {
  (void)in_sizes; (void)n_in; (void)out_size; (void)d_ws; (void)ws_size;
  const float* x0      = (const float*)d_in[0];
  const float* x1      = (const float*)d_in[1];
  const float* coeffs  = (const float*)d_in[2];
  const int*   idx0    = (const int*)d_in[3];
  const int*   idx1    = (const int*)d_in[4];
  const int*   idx_out = (const int*)d_in[5];
  float* out = (float*)d_out;
  tp_wmma_kernel<<<NBLOCKS, 256, 0, stream>>>(x0, x1, coeffs, idx0, idx1, idx_out, out);
}